// ResPNABlock_75771813036519
// MI455X (gfx1250) — compile-verified
//
#include <hip/hip_runtime.h>
#include <float.h>
#include <math.h>

// ResPNABlock for MI455X (gfx1250, wave32).
// - per-edge pre-GEMM decomposed into two N x 128 x 128 node GEMMs +
//   L2-resident edge gather with f32 atomics (B=25.6MB, AG=102MB < 192MB L2)
// - all GEMMs use V_WMMA_F32_16X16X4_F32 (exact fp32; GEMM FLOPs ~52G are not
//   the bottleneck, the atomic edge pass is)
// - 13F post-GEMM applies amp / 1/amp scalers at A-tile load time, never
//   materializing the N x 1664 concat (saves 333 MB)
// - GEMM waves compute a 16x64 strip (4 accumulators): A fragment loaded once
//   per K-step feeds 4 WMMAs; all fragment loads for a K-step are issued
//   before any WMMA so the backend can clause them and overlap WMMA execution
//   with outstanding loads (avoids s_wait_loadcnt 0 before every WMMA).

#define F 128
#define FS 512   // 4*F: aggr row stride (mean | min | max | sumsq->std)
#define NB 4     // N-tile blocking per wave (16 x 64 output strip)

typedef __attribute__((ext_vector_type(2))) float v2f;
typedef __attribute__((ext_vector_type(8))) float v8f;

__device__ __forceinline__ v8f wmma_f32(v2f a, v2f b, v8f c) {
  // D(16x16,f32) = A(16x4,f32) x B(4x16,f32) + C
  return __builtin_amdgcn_wmma_f32_16x16x4_f32(false, a, false, b, (short)0, c,
                                               false, false);
}

// ---------------------------------------------------------------- utilities
__global__ void k_zero_misc(int* __restrict__ deg, float* __restrict__ scal, int n) {
  int i = blockIdx.x * blockDim.x + threadIdx.x;
  if (i < n) deg[i] = 0;
  if (i == 0) scal[0] = 0.0f;
}

__global__ void k_deg(const int* __restrict__ dst, int* __restrict__ deg, int e) {
  int i = blockIdx.x * blockDim.x + threadIdx.x;
  if (i < e) atomicAdd(&deg[dst[i]], 1);
}

__global__ void k_avg_log(const int* __restrict__ deg, float* __restrict__ scal, int n) {
  __shared__ float red[256];
  float v = 0.0f;
  for (int i = blockIdx.x * blockDim.x + threadIdx.x; i < n;
       i += gridDim.x * blockDim.x)
    v += logf((float)deg[i] + 1.0f);
  red[threadIdx.x] = v;
  __syncthreads();
  for (int s = 128; s > 0; s >>= 1) {
    if ((int)threadIdx.x < s) red[threadIdx.x] += red[threadIdx.x + s];
    __syncthreads();
  }
  if (threadIdx.x == 0) atomicAdd(scal, red[0]);  // consumers divide by n
}

__global__ void k_init_aggr(float* __restrict__ AG, long long total) {
  long long i = (long long)blockIdx.x * blockDim.x + threadIdx.x;
  if (i >= total) return;
  int c = (int)(i & (FS - 1));
  float v = 0.0f;                                  // sum / sumsq slices
  if (c >= F && c < 2 * F) v = FLT_MAX;            // min slice
  else if (c >= 2 * F && c < 3 * F) v = -FLT_MAX;  // max slice
  AG[i] = v;
}

// ------------------------------------------------------- generic WMMA GEMM
// Y[M x Nout] = X[M x K] @ W^T + bias, W is [Nout x K] row-major (ld = ldw).
// One wave per 16x64 output strip (NB=4 column tiles sharing one A fragment).
// fp32 WMMA operand layout: lanes 0-15 carry K={0,1}, lanes 16-31 carry
// K={2,3} -> one float2 load per lane per fragment per K-step.
__global__ void k_gemm(const float* __restrict__ X, int ldx,
                       const float* __restrict__ W, int ldw,
                       const float* __restrict__ bias,
                       float* __restrict__ Y, int ldy,
                       int M, int K, int Nout) {
  int wave = (blockIdx.x * blockDim.x + threadIdx.x) >> 5;
  int lane = threadIdx.x & 31;
  int tn_cnt = Nout >> 6;  // strips of 64 columns
  int tm = wave / tn_cnt, ts = wave - tm * tn_cnt;
  if (tm * 16 >= M) return;
  int l = lane & 15;
  int koff = (lane >> 4) << 1;  // 0 for lanes 0-15, 2 for lanes 16-31
  const float* xp = X + (size_t)(tm * 16 + l) * ldx + koff;
  const float* wp[NB];
  v8f acc[NB];
#pragma unroll
  for (int t = 0; t < NB; ++t) {
    int tn = ts * NB + t;
    wp[t] = W + (size_t)(tn * 16 + l) * ldw + koff;
    float bv = bias ? bias[tn * 16 + l] : 0.0f;  // D column = f0 + (lane&15)
#pragma unroll
    for (int i = 0; i < 8; ++i) acc[t][i] = bv;
  }
  // K step of 8 = two independent fragment groups in flight per iteration.
  for (int k = 0; k < K; k += 8) {
    v2f a0 = *(const v2f*)(xp + k);
    v2f a1 = *(const v2f*)(xp + k + 4);
    v2f b0[NB], b1[NB];
#pragma unroll
    for (int t = 0; t < NB; ++t) b0[t] = *(const v2f*)(wp[t] + k);
#pragma unroll
    for (int t = 0; t < NB; ++t) b1[t] = *(const v2f*)(wp[t] + k + 4);
#pragma unroll
    for (int t = 0; t < NB; ++t) acc[t] = wmma_f32(a0, b0[t], acc[t]);
#pragma unroll
    for (int t = 0; t < NB; ++t) acc[t] = wmma_f32(a1, b1[t], acc[t]);
  }
  int rb = tm * 16 + ((lane >> 4) << 3);  // rows 0-7 (lo lanes) / 8-15 (hi)
#pragma unroll
  for (int t = 0; t < NB; ++t) {
    int col = (ts * NB + t) * 16 + l;
#pragma unroll
    for (int r = 0; r < 8; ++r) Y[(size_t)(rb + r) * ldy + col] = acc[t][r];
  }
}

// ------------------------------------- fused post GEMM over virtual K=13F
// Logical A row = [ h(128) | aggr(512) | aggr*amp(512) | aggr/amp(512) ],
// built on the fly at A-tile load; W is [F x 1664] row-major.
__global__ void k_gemm_post(const float* __restrict__ H,
                            const float* __restrict__ AG,
                            const float* __restrict__ amp,
                            const float* __restrict__ iamp,
                            const float* __restrict__ W,
                            const float* __restrict__ bias,
                            float* __restrict__ Y, int M) {
  const int K = 13 * F;  // 1664
  int wave = (blockIdx.x * blockDim.x + threadIdx.x) >> 5;
  int lane = threadIdx.x & 31;
  int tn_cnt = F >> 6;  // 2 strips of 64 columns
  int tm = wave / tn_cnt, ts = wave - tm * tn_cnt;
  if (tm * 16 >= M) return;
  int l = lane & 15;
  int koff = (lane >> 4) << 1;
  int row = tm * 16 + l;
  const float* hp = H + (size_t)row * F;
  const float* ap = AG + (size_t)row * FS;
  float ra = amp[row], ria = iamp[row];
  const float* wp[NB];
  v8f acc[NB];
#pragma unroll
  for (int t = 0; t < NB; ++t) {
    int tn = ts * NB + t;
    wp[t] = W + (size_t)(tn * 16 + l) * K + koff;
    float bv = bias[tn * 16 + l];
#pragma unroll
    for (int i = 0; i < 8; ++i) acc[t][i] = bv;
  }
  for (int k = 0; k < K; k += 4) {
    int kk = k + koff;  // 4-aligned quads never straddle 128-wide segments
    v2f a;
    if (kk < F) {
      a = *(const v2f*)(hp + kk);
    } else if (kk < F + FS) {
      a = *(const v2f*)(ap + (kk - F));
    } else if (kk < F + 2 * FS) {
      v2f t = *(const v2f*)(ap + (kk - F - FS));
      a = t * ra;
    } else {
      v2f t = *(const v2f*)(ap + (kk - F - 2 * FS));
      a = t * ria;
    }
    v2f b[NB];
#pragma unroll
    for (int t = 0; t < NB; ++t) b[t] = *(const v2f*)(wp[t] + k);
#pragma unroll
    for (int t = 0; t < NB; ++t) acc[t] = wmma_f32(a, b[t], acc[t]);
  }
  int rb = tm * 16 + ((lane >> 4) << 3);
#pragma unroll
  for (int t = 0; t < NB; ++t) {
    int col = (ts * NB + t) * 16 + l;
#pragma unroll
    for (int r = 0; r < 8; ++r) Y[(size_t)(rb + r) * F + col] = acc[t][r];
  }
}

// ------------------------------------------------- edge aggregation (L2-hot)
// One wave per edge; lane covers 4 channels (float4 gather of B[src]).
__global__ void k_edge_agg(const int* __restrict__ src, const int* __restrict__ dst,
                           const float* __restrict__ B, float* __restrict__ AG,
                           int E) {
  int t = blockIdx.x * blockDim.x + threadIdx.x;
  int e = t >> 5;
  if (e >= E) return;
  int lane = t & 31;
  int s = src[e], d = dst[e];
  const float4 v = *(const float4*)(B + (size_t)s * F + lane * 4);
  float* base = AG + (size_t)d * FS + lane * 4;
  float vv[4] = {v.x, v.y, v.z, v.w};
#pragma unroll
  for (int i = 0; i < 4; ++i) {
    atomicAdd(base + i, vv[i]);                  // sum(B)
    atomicMin(base + F + i, vv[i]);              // min(B)
    atomicMax(base + 2 * F + i, vv[i]);          // max(B)
    atomicAdd(base + 3 * F + i, vv[i] * vv[i]);  // sum(B^2)
  }
}

// ------------------------------------------- per-node aggregator finalize
// m_e = A[dst]+B[src]  =>  mean=(d*A+SB)/dc ; E[m^2]=(d*A^2+2A*SB+SB2)/dc ;
// max = A+maxB, min = A+minB (masked to 0 when deg==0, like the reference).
__global__ void k_finalize(const float* __restrict__ A, const int* __restrict__ deg,
                           const float* __restrict__ scal, float* __restrict__ AG,
                           float* __restrict__ amp, float* __restrict__ iamp,
                           int N) {
  int i = blockIdx.x * blockDim.x + threadIdx.x;
  int n = i >> 7, c = i & (F - 1);
  if (n >= N) return;
  float d = (float)deg[n];
  float dc = fmaxf(d, 1.0f);
  float* row = AG + (size_t)n * FS;
  float a = A[(size_t)n * F + c];
  float sb = row[c], mnv = row[F + c], mxv = row[2 * F + c], sq = row[3 * F + c];
  float mean = (d * a + sb) / dc;
  float e2 = (d * a * a + 2.0f * a * sb + sq) / dc;
  float var = fmaxf(e2 - mean * mean, 0.0f);
  float stdv = sqrtf(var + 1e-5f);
  bool has = d > 0.0f;
  row[c] = mean;
  row[F + c] = has ? (a + mnv) : 0.0f;
  row[2 * F + c] = has ? (a + mxv) : 0.0f;
  row[3 * F + c] = stdv;
  if (c == 0) {
    float adl = scal[0] / (float)N;  // avg_deg['log']
    float am = logf(dc + 1.0f) / adl;
    amp[n] = am;
    iamp[n] = 1.0f / am;
  }
}

// ---------------------------------------------------- BatchNorm (batch stats)
__global__ void k_bn_stats(const float* __restrict__ H, float* __restrict__ mu,
                           float* __restrict__ var, int N) {
  __shared__ float s1[256], s2[256];
  int c = blockIdx.x;  // one block per channel
  float a = 0.0f, b = 0.0f;
  for (int n = threadIdx.x; n < N; n += 256) {
    float v = H[(size_t)n * F + c];
    a += v;
    b += v * v;
  }
  s1[threadIdx.x] = a;
  s2[threadIdx.x] = b;
  __syncthreads();
  for (int s = 128; s > 0; s >>= 1) {
    if ((int)threadIdx.x < s) {
      s1[threadIdx.x] += s1[threadIdx.x + s];
      s2[threadIdx.x] += s2[threadIdx.x + s];
    }
    __syncthreads();
  }
  if (threadIdx.x == 0) {
    float m = s1[0] / (float)N;
    mu[c] = m;
    var[c] = s2[0] / (float)N - m * m;  // biased, like jnp .var()
  }
}

__global__ void k_bn_apply(const float* __restrict__ H, const float* __restrict__ mu,
                           const float* __restrict__ var, const float* __restrict__ g,
                           const float* __restrict__ be, const float* __restrict__ resid,
                           float* __restrict__ out, int total) {
  int i = blockIdx.x * blockDim.x + threadIdx.x;
  if (i >= total) return;
  int c = i & (F - 1);
  float v = g[c] * (H[i] - mu[c]) * rsqrtf(var[c] + 1e-5f) + be[c];
  v = fmaxf(v, 0.0f);
  if (resid) v = (v + resid[i]) * 0.70710678118654752f;  // (h + x)/sqrt(2)
  out[i] = v;
}

// ------------------------------------------------------------------ driver
extern "C" void kernel_launch(void* const* d_in, const int* in_sizes, int n_in,
                              void* d_out, int out_size, void* d_ws, size_t ws_size,
                              hipStream_t stream) {
  const float* x = (const float*)d_in[0];
  const int* ei = (const int*)d_in[1];
  const int N = in_sizes[0] / F;
  const int E = in_sizes[1] / 2;
  const int* srcv = ei;       // edge_index[0]
  const int* dstv = ei + E;   // edge_index[1]

  const float* Wpre[2] = {(const float*)d_in[2], (const float*)d_in[10]};
  const float* bpre[2] = {(const float*)d_in[3], (const float*)d_in[11]};
  const float* Wpost[2] = {(const float*)d_in[4], (const float*)d_in[12]};
  const float* bpost[2] = {(const float*)d_in[5], (const float*)d_in[13]};
  const float* Wlin[2] = {(const float*)d_in[6], (const float*)d_in[14]};
  const float* blin[2] = {(const float*)d_in[7], (const float*)d_in[15]};
  const float* gam[2] = {(const float*)d_in[8], (const float*)d_in[16]};
  const float* bet[2] = {(const float*)d_in[9], (const float*)d_in[17]};

  // workspace carve-up (~231 MB total)
  size_t NF = (size_t)N * F;
  float* ws = (float*)d_ws;
  float* A = ws;        ws += NF;       // h @ Wpre[:, :F]^T + b_pre   (x_i part)
  float* Bs = ws;       ws += NF;       // h @ Wpre[:, F:]^T           (x_j part)
  float* AG = ws;       ws += NF * 4;   // [sum->mean | min | max | sumsq->std]
  float* t0 = ws;       ws += NF;       // post_nn out
  float* t1 = ws;       ws += NF;       // lin out
  float* h = ws;        ws += NF;       // layer-1 output
  float* amp = ws;      ws += N;
  float* iamp = ws;     ws += N;
  float* mu = ws;       ws += F;
  float* var = ws;      ws += F;
  float* scal = ws;     ws += 1;
  int* deg = (int*)ws;

  dim3 blk(256);
  int gN = (int)((NF + 255) / 256);
  int gAgg = (int)((NF * 4 + 255) / 256);
  int gE32 = (int)(((size_t)E * 32 + 255) / 256);
  int tmcnt = (N + 15) / 16;
  // one wave per 16x64 output strip -> F/64 strips per row-tile
  int gGemm = (tmcnt * (F / 64) * 32 + 255) / 256;

  // degree histogram + avg_deg_log (shared by both layers)
  k_zero_misc<<<(N + 255) / 256, blk, 0, stream>>>(deg, scal, N);
  k_deg<<<(E + 255) / 256, blk, 0, stream>>>(dstv, deg, E);
  k_avg_log<<<256, blk, 0, stream>>>(deg, scal, N);

  for (int layer = 0; layer < 2; ++layer) {
    const float* hin = (layer == 0) ? x : h;
    k_init_aggr<<<gAgg, blk, 0, stream>>>(AG, (long long)NF * 4);
    // pre_nn split: A gets the x_i half (+bias), Bs the x_j half
    k_gemm<<<gGemm, blk, 0, stream>>>(hin, F, Wpre[layer], 2 * F, bpre[layer],
                                      A, F, N, F, F);
    k_gemm<<<gGemm, blk, 0, stream>>>(hin, F, Wpre[layer] + F, 2 * F, nullptr,
                                      Bs, F, N, F, F);
    k_edge_agg<<<gE32, blk, 0, stream>>>(srcv, dstv, Bs, AG, E);
    k_finalize<<<gN, blk, 0, stream>>>(A, deg, scal, AG, amp, iamp, N);
    k_gemm_post<<<gGemm, blk, 0, stream>>>(hin, AG, amp, iamp, Wpost[layer],
                                           bpost[layer], t0, N);
    k_gemm<<<gGemm, blk, 0, stream>>>(t0, F, Wlin[layer], F, blin[layer],
                                      t1, F, N, F, F);
    k_bn_stats<<<F, blk, 0, stream>>>(t1, mu, var, N);
    if (layer == 0)
      k_bn_apply<<<gN, blk, 0, stream>>>(t1, mu, var, gam[0], bet[0], nullptr,
                                         h, (int)NF);
    else
      k_bn_apply<<<gN, blk, 0, stream>>>(t1, mu, var, gam[1], bet[1], x,
                                         (float*)d_out, (int)NF);
  }
}